// TopologicalObserver_5068061409563
// MI455X (gfx1250) — compile-verified
//
#include <hip/hip_runtime.h>
#include <math.h>

typedef float v2f __attribute__((ext_vector_type(2)));
typedef float v8f __attribute__((ext_vector_type(8)));

#define D_MODEL 64

// cos/sin of theta_s = s * 2*pi/15, s = 0..15 (linspace(0, 2pi, 16), endpoint incl.)
__device__ __constant__ float kCosT[16] = {
     1.0000000f,  0.9135455f,  0.6691306f,  0.3090170f,
    -0.1045285f, -0.5000000f, -0.8090170f, -0.9781476f,
    -0.9781476f, -0.8090170f, -0.5000000f, -0.1045285f,
     0.3090170f,  0.6691306f,  0.9135455f,  1.0000000f };
__device__ __constant__ float kSinT[16] = {
     0.0000000f,  0.4067366f,  0.7431448f,  0.9510565f,
     0.9945219f,  0.8660254f,  0.5877853f,  0.2079117f,
    -0.2079117f, -0.5877853f, -0.8660254f, -0.9945219f,
    -0.9510565f, -0.7431448f, -0.4067366f,  0.0000000f };

// Per lane: accumulate the 16-step complex transmittance product for one pixel.
__device__ __forceinline__ void path_product(float dc, float cc, float rr,
                                             float& ar, float& ai) {
    ar = 1.0f; ai = 0.0f;
#pragma unroll
    for (int s = 0; s < 16; ++s) {
        float dp = fmaf(rr, kCosT[s], dc);      // delta_path
        float cp = fmaf(rr, kSinT[s], cc);      // chi_path
        float hd = 0.5f * dp;
        float ch = __cosf(hd);
        float sh = __sinf(hd);
        float cx = __cosf(cp);
        float sx = __sinf(cp);
        float re = ch * cx;                      // cos(d/2)cos(chi)
        float im = sh * (2.0f * sx * cx);        // sin(d/2)sin(2chi)
        float nr = fmaf(ar, re, -ai * im);
        float ni = fmaf(ar, im,  ai * re);
        ar = nr; ai = ni;
    }
}

// One wave handles 16 pixels: lanes 0..15 each own one pixel's accumulation,
// then V_WMMA_F32_16X16X4_F32 performs [re, im, 0, 0] x [W0; W1; 0; 0] + bias
// for four 16-wide feature tiles (64 features total).
__global__ __launch_bounds__(256) void topo_observer_kernel(
    const float* __restrict__ pp,    // (P, 3)
    const float* __restrict__ Wm,    // (2, 64) row-major
    const float* __restrict__ bias,  // (64)
    float* __restrict__ out,         // (P, 64)
    int P)
{
    const int lane = threadIdx.x & 31;
    const int nl   = lane & 15;
    const bool lo  = lane < 16;
    const long long gw = (long long)blockIdx.x * (blockDim.x >> 5) + (threadIdx.x >> 5);
    const long long p0 = gw * 16;

    if (p0 >= P) return;

    if (p0 + 16 <= (long long)P) {
        // ---- fast path: wave-uniform, EXEC all-ones through the WMMAs ----

        // Hoist all weight/bias loads unconditionally (every lane's address is
        // valid); their latency hides under the transcendental path product.
        float w0[4], w1[4], bv[4];
#pragma unroll
        for (int t = 0; t < 4; ++t) {
            const int n = t * 16 + nl;
            w0[t] = Wm[n];           // W row 0 (K=0)
            w1[t] = Wm[64 + n];      // W row 1 (K=1)
            bv[t] = bias[n];         // bias, depends only on feature column
        }

        const long long p = p0 + nl;                 // lanes 16..31 duplicate 0..15
        const float dc = pp[p * 3 + 0];
        const float cc = pp[p * 3 + 1];
        const float rr = pp[p * 3 + 2];
        float ar, ai;
        path_product(dc, cc, rr, ar, ai);

        // A-matrix (16x4 f32): VGPR0 = K0 (lanes 0-15) / K2 (lanes 16-31),
        //                      VGPR1 = K1 / K3. K2,K3 zero-padded.
        v2f A;
        A.x = lo ? ar : 0.0f;
        A.y = lo ? ai : 0.0f;

#pragma unroll
        for (int t = 0; t < 4; ++t) {
            const int n = t * 16 + nl;               // feature column 0..63
            // B-matrix (4x16 f32), mirrored layout: rows K=0,1 = W; K=2,3 = 0.
            v2f Bm;
            Bm.x = lo ? w0[t] : 0.0f;                // K=0 row / K=2 zero
            Bm.y = lo ? w1[t] : 0.0f;                // K=1 row / K=3 zero
            const float b = bv[t];
            v8f C = { b, b, b, b, b, b, b, b };

            v8f D = __builtin_amdgcn_wmma_f32_16x16x4_f32(
                false, A, false, Bm, (short)0, C, false, false);

            // D layout: VGPR i -> row (i | i+8), col = lane%16.
            const long long rbase = p0 + (lo ? 0 : 8);
#pragma unroll
            for (int i = 0; i < 8; ++i) {
                out[(rbase + i) * D_MODEL + n] = D[i];
            }
        }
    } else {
        // ---- scalar tail (unreachable for the reference shape) ----
        const long long p = p0 + lane;
        if (p < P) {
            const float dc = pp[p * 3 + 0];
            const float cc = pp[p * 3 + 1];
            const float rr = pp[p * 3 + 2];
            float ar, ai;
            path_product(dc, cc, rr, ar, ai);
#pragma unroll 4
            for (int j = 0; j < D_MODEL; ++j) {
                out[p * D_MODEL + j] = fmaf(ar, Wm[j], fmaf(ai, Wm[64 + j], bias[j]));
            }
        }
    }
}

extern "C" void kernel_launch(void* const* d_in, const int* in_sizes, int n_in,
                              void* d_out, int out_size, void* d_ws, size_t ws_size,
                              hipStream_t stream) {
    const float* pp   = (const float*)d_in[0];  // path_params_grid (B,H,W,3)
    const float* Wm   = (const float*)d_in[1];  // (2, 64)
    const float* bias = (const float*)d_in[2];  // (64,)
    float* out        = (float*)d_out;          // (B,H,W,64)

    const int P = in_sizes[0] / 3;              // total pixels = B*H*W
    const int waves_needed  = (P + 15) / 16;
    const int waves_per_blk = 8;                // 256 threads
    const int blocks = (waves_needed + waves_per_blk - 1) / waves_per_blk;

    topo_observer_kernel<<<blocks, 256, 0, stream>>>(pp, Wm, bias, out, P);
}